// RadialBasisExpansion_22265110462974
// MI455X (gfx1250) — compile-verified
//
#include <hip/hip_runtime.h>

// CDNA5 / gfx1250, wave32. RBF edge expansion (bandwidth-bound: ~0.8 GB traffic,
// ~35us floor at 23.3 TB/s).
// Exponent tile via V_WMMA_F32_16X16X4_F32:
//   A row (per edge m):      [ d_m^2,   d_m,            1,             0 ]
//   B col (per gaussian g):  [ spacing, -2*spacing*c_g, spacing*c_g^2, 0 ]
//   D[m][g] = spacing*(d_m - c_g)^2  ->  exp() -> rbf weight.
// Each wave stages its 32x50 fp32 output tile contiguously in LDS, then drains
// it with GLOBAL_STORE_ASYNC_FROM_LDS_B128/B64 (ASYNCcnt async-DMA path).

typedef __attribute__((ext_vector_type(2))) float v2f;
typedef __attribute__((ext_vector_type(8))) float v8f;

// Async-store builtin parameter types (confirmed by hipcc diagnostics):
typedef int v4i __attribute__((vector_size(16)));
typedef int v2i __attribute__((vector_size(8)));

#define N_GAUSS 50
#define WAVES_PER_BLOCK 8
#define TILE_FLOATS (32 * N_GAUSS)   // 1600 fp32 per wave tile (6400B)

#if __has_builtin(__builtin_amdgcn_global_store_async_from_lds_b128) && \
    __has_builtin(__builtin_amdgcn_global_store_async_from_lds_b64)
#define USE_ASYNC_STORE 1
#else
#define USE_ASYNC_STORE 0
#endif

__global__ __launch_bounds__(256) void rbf_wmma_kernel(
    const float* __restrict__ pos,      // [N,3]
    const int*   __restrict__ edges,    // [2,E]
    const float* __restrict__ centers,  // [50]
    float* __restrict__ out_edges,      // [2,E] as float (ids < 2^24: exact)
    float* __restrict__ out_dist,       // [E]
    float* __restrict__ out_rbf,        // [E,50]
    long long E)
{
    __shared__ __align__(16) float stage[WAVES_PER_BLOCK * TILE_FLOATS];

    const int lane = threadIdx.x & 31;
    const int wave = threadIdx.x >> 5;
    const long long waveBase =
        ((long long)blockIdx.x * WAVES_PER_BLOCK + wave) * 32;
    const long long e  = waveBase + lane;
    const bool valid   = (e < E);
    const long long ec = valid ? e : (E - 1);   // select, keeps EXEC full for loads

    // ---- per-lane distance (32 edges per wave) ----
    const int s = edges[ec];
    const int t = edges[E + ec];
    const float dx = pos[3*s + 0] - pos[3*t + 0];
    const float dy = pos[3*s + 1] - pos[3*t + 1];
    const float dz = pos[3*s + 2] - pos[3*t + 2];
    const float dist = sqrtf(dx*dx + dy*dy + dz*dz);

    if (valid) {
        const unsigned ue = (unsigned)e;        // E,2E < 2^31: 32-bit indexing
        out_dist[ue]                 = dist;
        out_edges[ue]                = (float)s;
        out_edges[(unsigned)E + ue]  = (float)t;
    }

    // ---- spacing from centers ----
    const float dcen    = centers[1] - centers[0];
    const float spacing = -0.5f / (dcen * dcen);

    // ---- B tiles: 4 gaussian tiles of 16 columns ----
    // 32-bit B (4x16): V0 lanes0-15 = K=0 row, lanes16-31 = K=2 row;
    //                  V1 lanes0-15 = K=1 row, lanes16-31 = K=3 row.
    const int  n  = lane & 15;
    const bool lo = (lane < 16);
    v2f b[4];
#pragma unroll
    for (int tg = 0; tg < 4; ++tg) {
        const int g   = tg * 16 + n;
        const float c = centers[g < N_GAUSS ? g : (N_GAUSS - 1)];
        b[tg].x = lo ? spacing              : spacing * c * c;  // K=0 | K=2
        b[tg].y = lo ? (-2.0f * spacing * c) : 0.0f;            // K=1 | K=3
    }

    const int m0 = lo ? 0 : 8;                  // D row base for this half-wave

    // ---- 2 edge-tiles x 4 gaussian-tiles = 8 WMMAs per wave ----
    // 32-bit A (16x4): lanes0-15 row M=lane: V0=K0, V1=K1;
    //                  lanes16-31 row M=lane-16: V0=K2, V1=K3.
    // All results staged to LDS unconditionally; out-of-range rows of partial
    // waves are simply never drained.
#pragma unroll
    for (int et = 0; et < 2; ++et) {
        const float dm = __shfl(dist, et * 16 + n, 32);
        v2f a;
        a.x = lo ? dm * dm : 1.0f;   // K0 | K2
        a.y = lo ? dm      : 0.0f;   // K1 | K3

#pragma unroll
        for (int tg = 0; tg < 4; ++tg) {
            v8f acc = {};
            acc = __builtin_amdgcn_wmma_f32_16x16x4_f32(
                false, a, false, b[tg], (short)0, acc, false, false);

            float w[8];
#pragma unroll
            for (int j = 0; j < 8; ++j)
                w[j] = __expf(acc[j]);           // v_exp_f32, before any masking

            // Column guard is compile-time for tg<3 (g = tg*16+n <= 47 < 50);
            // only the tg==3 tile has a (tiny) divergent region: n<2.
            if (tg < 3 || n < 2) {
                const int g = tg * 16 + n;
                float* lp = &stage[wave * TILE_FLOATS +
                                   (et * 16 + m0) * N_GAUSS + g];
#pragma unroll
                for (int j = 0; j < 8; ++j)
                    lp[j * N_GAUSS] = w[j];      // ds_store_2addr_b32 pairs
            }
        }
    }

    // ---- drain the wave's contiguous 6400B tile: LDS -> global ----
    asm volatile("s_wait_dscnt 0x0" ::: "memory");  // LDS writes visible below
    float* lsrc = &stage[wave * TILE_FLOATS];
    float* gdst = out_rbf + (size_t)waveBase * N_GAUSS;   // 6400B-aligned

#if USE_ASYNC_STORE
    if (waveBase + 32 <= E) {                    // uniform branch; hot path
#pragma unroll
        for (int k = 0; k < 12; ++k) {           // 12 x (32 lanes x 16B) = 1536 fp32
            __builtin_amdgcn_global_store_async_from_lds_b128(
                (v4i*)(gdst + k * 128 + lane * 4),
                (v4i*)(lsrc + k * 128 + lane * 4), 0, 0);
        }
        __builtin_amdgcn_global_store_async_from_lds_b64(  // tail 64 fp32
            (v2i*)(gdst + 1536 + lane * 2),
            (v2i*)(lsrc + 1536 + lane * 2), 0, 0);
        // completion covered by implicit wait-idle at S_ENDPGM
    } else
#endif
    {
        // Cold tail (only when E % 32 != 0 / partial wave): compact copy loop.
        const long long rem = E - waveBase;
        const int count = rem <= 0 ? 0 : (rem >= 32 ? 32 : (int)rem);
        const int total = count * N_GAUSS;
        for (int idx = lane; idx < total; idx += 32)
            gdst[idx] = lsrc[idx];
    }
}

extern "C" void kernel_launch(void* const* d_in, const int* in_sizes, int n_in,
                              void* d_out, int out_size, void* d_ws, size_t ws_size,
                              hipStream_t stream) {
    const float* pos     = (const float*)d_in[0];
    const int*   edges   = (const int*)d_in[1];
    const float* centers = (const float*)d_in[2];

    const long long E = (long long)in_sizes[1] / 2;

    // Output layout (tuple flattened, fp32): [edges 2E][dist E][rbf E*50]
    float* out_edges = (float*)d_out;
    float* out_dist  = out_edges + 2 * E;
    float* out_rbf   = out_dist + E;

    const int block = 256;                       // 8 wave32 waves / block
    const long long edgesPerBlock = 256;         // 32 edges per wave
    const int grid = (int)((E + edgesPerBlock - 1) / edgesPerBlock);

    rbf_wmma_kernel<<<grid, block, 0, stream>>>(
        pos, edges, centers, out_edges, out_dist, out_rbf, E);
}